// VanillaGCNEncoder_13168369729823
// MI455X (gfx1250) — compile-verified
//
#include <hip/hip_runtime.h>
#include <hip/hip_bf16.h>
#include <math.h>
#include <stdint.h>

typedef float v2f __attribute__((ext_vector_type(2)));
typedef float v8f __attribute__((ext_vector_type(8)));

#define N_NODES 100000
#define E_EDGES 3200000
#define IN_CH   128
#define GCN_CH  256
#define OUT_CH  64

#define KCHUNK     128          // K rows staged in LDS per pass
#define LDS_STRIDE 80           // floats per staged row (64 data + 16 pad -> conflict-free)

// ---------------------------------------------------------------------------
// gcn_norm: deg[i] = 1 (self loop) + sum_{e: col[e]==i} ew[e];  dis = rsqrt(deg)
// ---------------------------------------------------------------------------
__global__ void k_init_deg(float* __restrict__ deg, int n) {
  int i = blockIdx.x * blockDim.x + threadIdx.x;
  if (i < n) deg[i] = 1.0f;  // self-loop weight 1
}

__global__ void k_accum_deg(const long long* __restrict__ col,
                            const float* __restrict__ ew,
                            float* __restrict__ deg, int e) {
  int i = blockIdx.x * blockDim.x + threadIdx.x;
  if (i < e) atomicAdd(&deg[col[i]], ew[i]);
}

__global__ void k_rsqrt_inplace(float* __restrict__ deg, int n) {
  int i = blockIdx.x * blockDim.x + threadIdx.x;
  if (i < n) {
    float d = deg[i];
    deg[i] = (d > 0.0f) ? rsqrtf(d) : 0.0f;
  }
}

__global__ void k_edge_norm(const long long* __restrict__ row,
                            const long long* __restrict__ col,
                            const float* __restrict__ ew,
                            const float* __restrict__ dis,
                            float* __restrict__ norm, int e) {
  int i = blockIdx.x * blockDim.x + threadIdx.x;
  if (i < e) norm[i] = dis[row[i]] * ew[i] * dis[col[i]];
}

// ---------------------------------------------------------------------------
// f32 WMMA GEMM with async-LDS staging of the B (weight) panel.
//
// Block = 256 threads (8 waves) computes a 128-row x 64-col output tile:
//   wave w -> m-tile (mb*8 + w), 4 n-tiles (4 accumulators, A frag reused 4x).
// Per K-chunk (128 rows): the block stages B[kc:kc+128, n0:n0+64] into LDS via
// global_load_async_to_lds_b128 (ASYNCcnt), waits with s_wait_asynccnt, then
// runs V_WMMA_F32_16X16X4_F32 with B fragments from ds_load (stride-80 rows
// put lanes 0-15 / 16-31 in disjoint LDS bank groups -> conflict free).
//
// A frag (16x4 f32): lanes 0-15 -> M=lane, K=k+{0,1}; lanes 16-31 -> K=k+{2,3}.
// C/D (16x16 f32): vgpr r -> M = r (lanes 0-15) / r+8 (lanes 16-31), N=lane&15.
// M-guard is wave-uniform; all waves join the cooperative load + barriers.
// mode: 0 = plain store, 1 = tanh(acc + bias[n]) epilogue.
// ---------------------------------------------------------------------------
__global__ void k_gemm_wmma_lds(const float* __restrict__ A,
                                const float* __restrict__ B,
                                const float* __restrict__ bias,
                                float* __restrict__ C,
                                int M, int K, int Ncols, int mBlocks, int mode) {
  __shared__ float Bs[KCHUNK * LDS_STRIDE];   // 40 KB

  const int lane = threadIdx.x & 31;
  const int wave = threadIdx.x >> 5;
  const int mb   = blockIdx.x % mBlocks;
  const int nb   = blockIdx.x / mBlocks;
  const int n0   = nb * 64;
  const int mtile = mb * 8 + wave;
  const bool valid = (mtile * 16 < M);        // wave-uniform

  const int koff = (lane >> 4) << 1;          // 0 (lanes 0-15) / 2 (lanes 16-31)
  const int nlo  = lane & 15;

  v8f acc[4];
#pragma unroll
  for (int t = 0; t < 4; ++t) acc[t] = (v8f){0.f,0.f,0.f,0.f,0.f,0.f,0.f,0.f};

  const uint32_t ldsBase = (uint32_t)(uintptr_t)&Bs[0];
  const float* arow = valid ? (A + (size_t)(mtile * 16 + nlo) * K + koff) : A;

  for (int kc = 0; kc < K; kc += KCHUNK) {
    // ---- cooperative async stage of B[kc:kc+128, n0:n0+64] into LDS ----
    for (int idx = threadIdx.x; idx < KCHUNK * 16; idx += 256) {
      const int r   = idx >> 4;               // row within chunk
      const int seg = idx & 15;               // 16-byte segment (4 floats)
      const uint32_t loff = ldsBase + (uint32_t)(r * (LDS_STRIDE * 4) + seg * 16);
      const float* g = B + (size_t)(kc + r) * Ncols + n0 + seg * 4;
      asm volatile("global_load_async_to_lds_b128 %0, %1, off"
                   :: "v"(loff), "v"(g) : "memory");
    }
    asm volatile("s_wait_asynccnt 0x0" ::: "memory");
    __syncthreads();

    if (valid) {
      for (int k = 0; k < KCHUNK; k += 4) {
        v2f a = *(const v2f*)(arow + kc + k);             // 8B-aligned pair
        const float* b0 = &Bs[(k + koff) * LDS_STRIDE + nlo];
#pragma unroll
        for (int t = 0; t < 4; ++t) {
          v2f b;
          b.x = b0[t * 16];                               // row k+koff
          b.y = b0[LDS_STRIDE + t * 16];                  // row k+koff+1
          acc[t] = __builtin_amdgcn_wmma_f32_16x16x4_f32(
              false, a, false, b, (short)0, acc[t], false, false);
        }
      }
    }
    __syncthreads();                                      // before next overwrite
  }

  if (valid) {
    const int mC = mtile * 16 + ((lane >> 4) << 3);
#pragma unroll
    for (int t = 0; t < 4; ++t) {
      const int n = n0 + t * 16 + nlo;
      float* cp = C + (size_t)mC * Ncols + n;
      if (mode == 1) {
        const float bv = bias[n];
#pragma unroll
        for (int r = 0; r < 8; ++r) cp[(size_t)r * Ncols] = tanhf(acc[t][r] + bv);
      } else {
#pragma unroll
        for (int r = 0; r < 8; ++r) cp[(size_t)r * Ncols] = acc[t][r];
      }
    }
  }
}

// ---------------------------------------------------------------------------
// Self-loop seed: hB[i,:] = dis[i]^2 * hA[i,:]
// ---------------------------------------------------------------------------
__global__ void k_selfloop(const float* __restrict__ h,
                           const float* __restrict__ dis,
                           float* __restrict__ out, int total) {
  int i = blockIdx.x * blockDim.x + threadIdx.x;
  if (i < total) {
    float d = dis[i >> 8];                  // i / 256
    out[i] = d * d * h[i];
  }
}

// ---------------------------------------------------------------------------
// Edge scatter: one wave per edge. 32 lanes x 2 float4 = 256 channels.
// Coalesced 512B gather per step from h[row], f32 atomics into out[col].
// ---------------------------------------------------------------------------
__global__ void k_scatter(const float* __restrict__ h,
                          const long long* __restrict__ row,
                          const long long* __restrict__ col,
                          const float* __restrict__ norm,
                          float* __restrict__ out, int e) {
  const int wid  = (blockIdx.x * blockDim.x + threadIdx.x) >> 5;
  const int lane = threadIdx.x & 31;
  if (wid >= e) return;                     // wave-uniform
  const long long r = row[wid];
  const long long c = col[wid];
  const float w = norm[wid];
  const float4* __restrict__ hp = (const float4*)(h + (size_t)r * GCN_CH);
  float* __restrict__ op = out + (size_t)c * GCN_CH;
#pragma unroll
  for (int j = 0; j < 2; ++j) {
    const int q = lane + 32 * j;
    float4 v = hp[q];
    atomicAdd(op + 4 * q + 0, w * v.x);
    atomicAdd(op + 4 * q + 1, w * v.y);
    atomicAdd(op + 4 * q + 2, w * v.z);
    atomicAdd(op + 4 * q + 3, w * v.w);
  }
}

// ---------------------------------------------------------------------------
// Fused bias + ReLU over [N, 256]
// ---------------------------------------------------------------------------
__global__ void k_bias_relu(float* __restrict__ h,
                            const float* __restrict__ bias, int total) {
  int i = blockIdx.x * blockDim.x + threadIdx.x;
  if (i < total) {
    float v = h[i] + bias[i & (GCN_CH - 1)];
    h[i] = v > 0.0f ? v : 0.0f;
  }
}

// ---------------------------------------------------------------------------
extern "C" void kernel_launch(void* const* d_in, const int* in_sizes, int n_in,
                              void* d_out, int out_size, void* d_ws, size_t ws_size,
                              hipStream_t stream) {
  const float*     x   = (const float*)d_in[0];
  const long long* ei  = (const long long*)d_in[1];   // int64 [2,E]
  const float*     ew  = (const float*)d_in[2];
  const float*     W1  = (const float*)d_in[3];
  const float*     b1  = (const float*)d_in[4];
  const float*     W2  = (const float*)d_in[5];
  const float*     b2  = (const float*)d_in[6];
  const float*     W3  = (const float*)d_in[7];
  const float*     b3  = (const float*)d_in[8];
  float*           out = (float*)d_out;

  const long long* row = ei;              // edge_index[0]
  const long long* col = ei + E_EDGES;    // edge_index[1]

  // Workspace carve-up (all offsets stay 16B-aligned):
  float* dis  = (float*)d_ws;                         // N
  float* norm = dis + N_NODES;                        // E
  float* hA   = norm + E_EDGES;                       // N*256
  float* hB   = hA + (size_t)N_NODES * GCN_CH;        // N*256

  const int T = 256;
  const int nN  = (N_NODES + T - 1) / T;
  const int nE  = (E_EDGES + T - 1) / T;
  const int nNC = (N_NODES * GCN_CH + T - 1) / T;
  const int scatterBlocks = (E_EDGES + 7) / 8;        // 8 waves/block, wave/edge

  const int mTiles  = N_NODES / 16;                   // 6250
  const int mBlocks = (mTiles + 7) / 8;               // 782 (last block ragged)
  const int gridBig = mBlocks * (GCN_CH / 64);        // 782 * 4
  const int gridOut = mBlocks * (OUT_CH / 64);        // 782 * 1

  // --- normalization ---
  k_init_deg<<<nN, T, 0, stream>>>(dis, N_NODES);
  k_accum_deg<<<nE, T, 0, stream>>>(col, ew, dis, E_EDGES);
  k_rsqrt_inplace<<<nN, T, 0, stream>>>(dis, N_NODES);
  k_edge_norm<<<nE, T, 0, stream>>>(row, col, ew, dis, norm, E_EDGES);

  // --- layer 1: hA = x @ W1; hB = selfloop + scatter(norm * hA[row]); relu(+b1) ---
  k_gemm_wmma_lds<<<gridBig, T, 0, stream>>>(x, W1, nullptr, hA,
                                             N_NODES, IN_CH, GCN_CH, mBlocks, 0);
  k_selfloop<<<nNC, T, 0, stream>>>(hA, dis, hB, N_NODES * GCN_CH);
  k_scatter<<<scatterBlocks, T, 0, stream>>>(hA, row, col, norm, hB, E_EDGES);
  k_bias_relu<<<nNC, T, 0, stream>>>(hB, b1, N_NODES * GCN_CH);

  // --- layer 2 ---
  k_gemm_wmma_lds<<<gridBig, T, 0, stream>>>(hB, W2, nullptr, hA,
                                             N_NODES, GCN_CH, GCN_CH, mBlocks, 0);
  k_selfloop<<<nNC, T, 0, stream>>>(hA, dis, hB, N_NODES * GCN_CH);
  k_scatter<<<scatterBlocks, T, 0, stream>>>(hA, row, col, norm, hB, E_EDGES);
  k_bias_relu<<<nNC, T, 0, stream>>>(hB, b2, N_NODES * GCN_CH);

  // --- output: out = tanh(hB @ W3 + b3) ---
  k_gemm_wmma_lds<<<gridOut, T, 0, stream>>>(hB, W3, b3, out,
                                             N_NODES, GCN_CH, OUT_CH, mBlocks, 1);
}